// Attn_20048907338076
// MI455X (gfx1250) — compile-verified
//
#include <hip/hip_runtime.h>
#include <hip/hip_bf16.h>
#include <math.h>

typedef float v2f __attribute__((ext_vector_type(2)));
typedef float v8f __attribute__((ext_vector_type(8)));
typedef int v4i_vs __attribute__((vector_size(16)));  // matches builtin proto

#define AS1 __attribute__((address_space(1)))
#define AS3 __attribute__((address_space(3)))

#define WMMA_F32(a, b, c) \
  __builtin_amdgcn_wmma_f32_16x16x4_f32(false, (a), false, (b), (short)0, (c), false, false)

static __device__ __forceinline__ v8f vzero8() {
  v8f z = {0.f, 0.f, 0.f, 0.f, 0.f, 0.f, 0.f, 0.f};
  return z;
}

// 16-byte async global->LDS copy (per-lane). Falls back to a synchronous copy
// if the toolchain lacks the gfx1250 async builtins.
static __device__ __forceinline__ void async_copy16(const float* g, float* l) {
#if __has_builtin(__builtin_amdgcn_global_load_async_to_lds_b128)
  __builtin_amdgcn_global_load_async_to_lds_b128((AS1 v4i_vs*)g,
                                                 (AS3 v4i_vs*)l, 0, 0);
#else
  *(float4*)l = *(const float4*)g;
#endif
}

static __device__ __forceinline__ void async_wait_all() {
#if __has_builtin(__builtin_amdgcn_s_wait_asynccnt)
  __builtin_amdgcn_s_wait_asynccnt(0);
#endif
}

// ---------------------------------------------------------------------------
// GEMM: C[M,N] = A[M,K]*B[K,N] + bias[N]; mode==1 applies floor(x/8) epilogue.
// 128 threads / 4 waves, 64x64 tile, wave tile 32x32, double-buffered LDS with
// async global->LDS staging overlapping the WMMA work.
// ---------------------------------------------------------------------------
__global__ __launch_bounds__(128) void gemm_bias_wmma_f32(
    const float* __restrict__ A, const float* __restrict__ Bw,
    const float* __restrict__ bias, float* __restrict__ C,
    int M, int N, int K, int mode) {
  __shared__ __align__(16) float Asl[2][64][20];  // 80B rows (16B aligned)
  __shared__ __align__(16) float Bsl[2][16][68];  // 272B rows (16B aligned)

  const int t = threadIdx.x;
  const int lane = t & 31;
  const int wv = t >> 5;
  const int l15 = lane & 15;
  const int lhi = lane >> 4;

  const int m0 = blockIdx.y * 64;
  const int n0 = blockIdx.x * 64;
  const int wm = (wv >> 1) * 32;
  const int wn = (wv & 1) * 32;

  v8f acc[2][2];
  for (int i = 0; i < 2; ++i)
    for (int j = 0; j < 2; ++j) acc[i][j] = vzero8();

  auto issue_tiles = [&](int k0, int buf) {
#pragma unroll
    for (int i = 0; i < 2; ++i) {
      int c = t + i * 128;      // 0..255
      int row = c >> 2;         // 0..63
      int c4 = (c & 3) * 4;     // 0,4,8,12
      async_copy16(&A[(size_t)(m0 + row) * K + k0 + c4], &Asl[buf][row][c4]);
    }
#pragma unroll
    for (int i = 0; i < 2; ++i) {
      int c = t + i * 128;
      int row = c >> 4;         // 0..15
      int c16 = (c & 15) * 4;   // 0..60
      async_copy16(&Bw[(size_t)(k0 + row) * N + n0 + c16], &Bsl[buf][row][c16]);
    }
  };

  issue_tiles(0, 0);
  const int niter = K >> 4;
  for (int it = 0; it < niter; ++it) {
    const int cur = it & 1;
    async_wait_all();      // current tile's async copies complete
    __syncthreads();       // visible to (and past compute of) all waves
    if (it + 1 < niter) issue_tiles((it + 1) * 16, cur ^ 1);

#pragma unroll
    for (int kk = 0; kk < 16; kk += 4) {
      v2f a[2], b[2];
#pragma unroll
      for (int i = 0; i < 2; ++i)
        a[i] = *(v2f*)&Asl[cur][wm + i * 16 + l15][kk + lhi * 2];
#pragma unroll
      for (int j = 0; j < 2; ++j) {
        int col = wn + j * 16 + l15;
        b[j].x = Bsl[cur][kk + lhi * 2][col];
        b[j].y = Bsl[cur][kk + 1 + lhi * 2][col];
      }
#pragma unroll
      for (int i = 0; i < 2; ++i)
#pragma unroll
        for (int j = 0; j < 2; ++j) acc[i][j] = WMMA_F32(a[i], b[j], acc[i][j]);
    }
  }

  // Epilogue: C-layout -> lane<16: rows v, col=lane; lane>=16: rows v+8
#pragma unroll
  for (int i = 0; i < 2; ++i)
#pragma unroll
    for (int j = 0; j < 2; ++j) {
      int n = n0 + wn + j * 16 + l15;
      float bb = bias[n];
#pragma unroll
      for (int v = 0; v < 8; ++v) {
        int m = m0 + wm + i * 16 + v + lhi * 8;
        float val = acc[i][j][v] + bb;
        if (mode == 1) val = floorf(val * 0.125f);  // q // sqrt(Dh)
        C[(size_t)m * N + n] = val;
      }
    }
}

// ---------------------------------------------------------------------------
// Flash attention: per (b,h), 64 queries/block (4 waves x 16 queries).
// S^T = K*Q^T (softmax reductions per-lane + one shfl_xor(16)),
// O^T = V^T*P^T.  K/V tiles double-buffered in a 64KB LDS arena with a
// 16B-chunk XOR swizzle; staged with async copies overlapping compute.
// ---------------------------------------------------------------------------
__global__ __launch_bounds__(128) void flash_attn_wmma_f32(
    const float* __restrict__ Qw, const float* __restrict__ Kw,
    const float* __restrict__ Vw, float* __restrict__ Ow,
    int Lseq, int Hn) {
  constexpr int TILE_F = 64 * 64;
  __shared__ __align__(16) float smem[4 * TILE_F];  // [K0, V0, K1, V1] = 64KB

  const int t = threadIdx.x;
  const int lane = t & 31;
  const int wv = t >> 5;
  const int l15 = lane & 15;
  const int lhi = lane >> 4;

  const int q0 = blockIdx.x * 64;
  const int h = blockIdx.y;
  const int b = blockIdx.z;

  auto swz = [](int row, int col) {  // 16B-chunk XOR swizzle, row stride 64
    return row * 64 + ((((col >> 2) ^ (row & 15)) << 2) | (col & 3));
  };

  // Stage Q tile through the K1 region (first overwritten by tile-1 prefetch,
  // which is only issued after a barrier that follows fragment extraction).
#pragma unroll
  for (int i = 0; i < 8; ++i) {
    int c = t + i * 128;  // 0..1023
    int row = c >> 4;
    int ch = c & 15;
    size_t g = (((size_t)b * Lseq + (q0 + row)) * Hn + h) * 64 + ch * 4;
    async_copy16(&Qw[g], &smem[2 * TILE_F + row * 64 + ((ch ^ (row & 15)) << 2)]);
  }
  async_wait_all();
  __syncthreads();

  // Per-wave Q^T B-fragments (resident for the whole key loop)
  v2f Bq[16];
  const int qrow = wv * 16 + l15;
#pragma unroll
  for (int s = 0; s < 16; ++s) {
    int idx = qrow * 64 + ((s ^ (qrow & 15)) << 2) + lhi * 2;
    Bq[s] = *(v2f*)&smem[2 * TILE_F + idx];
  }

  auto issue_kv = [&](int jt, int buf) {
    float* kr = &smem[buf * 2 * TILE_F];
    float* vr = &smem[buf * 2 * TILE_F + TILE_F];
#pragma unroll
    for (int i = 0; i < 8; ++i) {
      int c = t + i * 128;
      int row = c >> 4;
      int ch = c & 15;
      size_t g = (((size_t)b * Lseq + (jt * 64 + row)) * Hn + h) * 64 + ch * 4;
      int l = row * 64 + ((ch ^ (row & 15)) << 2);
      async_copy16(&Kw[g], &kr[l]);
      async_copy16(&Vw[g], &vr[l]);
    }
  };

  float m_run = -3.0e38f;
  float l_run = 0.0f;
  v8f o_acc[4];
  for (int d = 0; d < 4; ++d) o_acc[d] = vzero8();

  const int qglob = q0 + wv * 16 + l15;
  const int nkt = q0 / 64 + 1;  // causal: key tiles up to the diagonal

  issue_kv(0, 0);
  for (int jt = 0; jt < nkt; ++jt) {
    const int cur = jt & 1;
    async_wait_all();
    __syncthreads();
    if (jt + 1 < nkt) issue_kv(jt + 1, cur ^ 1);

    const float* Ksb = &smem[cur * 2 * TILE_F];
    const float* Vsb = &smem[cur * 2 * TILE_F + TILE_F];

    // S^T tile: 4 key-subtiles of 16 x (16 queries)
    v8f sacc[4];
    for (int sub = 0; sub < 4; ++sub) sacc[sub] = vzero8();
#pragma unroll
    for (int kk = 0; kk < 16; ++kk) {
#pragma unroll
      for (int sub = 0; sub < 4; ++sub) {
        int row = sub * 16 + l15;
        int idx = row * 64 + ((kk ^ (row & 15)) << 2) + lhi * 2;
        v2f a = *(v2f*)&Ksb[idx];
        sacc[sub] = WMMA_F32(a, Bq[kk], sacc[sub]);
      }
    }

    // Causal mask (diagonal tile only) + online softmax
    float p[4][8];
    float tmax = -3.0e38f;
    const bool diag = (jt == nkt - 1);
#pragma unroll
    for (int sub = 0; sub < 4; ++sub)
#pragma unroll
      for (int v = 0; v < 8; ++v) {
        float s = sacc[sub][v];
        int kglob = jt * 64 + sub * 16 + v + lhi * 8;
        if (diag && kglob > qglob) s = -3.0e38f;
        p[sub][v] = s;
        tmax = fmaxf(tmax, s);
      }
    tmax = fmaxf(tmax, __shfl_xor(tmax, 16, 32));
    float m_new = fmaxf(m_run, tmax);
    float corr = __expf(m_run - m_new);
    float tsum = 0.0f;
#pragma unroll
    for (int sub = 0; sub < 4; ++sub)
#pragma unroll
      for (int v = 0; v < 8; ++v) {
        float e = __expf(p[sub][v] - m_new);
        p[sub][v] = e;
        tsum += e;
      }
    tsum += __shfl_xor(tsum, 16, 32);
    l_run = l_run * corr + tsum;
    m_run = m_new;
#pragma unroll
    for (int d = 0; d < 4; ++d)
#pragma unroll
      for (int v = 0; v < 8; ++v) o_acc[d][v] *= corr;

    // PV: O^T += V^T * P^T; B-fragments of P^T rebuilt with half-wave swaps.
#pragma unroll
    for (int kc4 = 0; kc4 < 16; ++kc4) {
      const int sub = kc4 >> 2;
      const int kc = (kc4 & 3) * 4;
      v2f bf;
      if (kc < 8) {
        float t0 = __shfl_xor(p[sub][kc + 2], 16, 32);
        float t1 = __shfl_xor(p[sub][kc + 3], 16, 32);
        bf.x = (lane < 16) ? p[sub][kc] : t0;
        bf.y = (lane < 16) ? p[sub][kc + 1] : t1;
      } else {
        float t0 = __shfl_xor(p[sub][kc - 8], 16, 32);
        float t1 = __shfl_xor(p[sub][kc - 7], 16, 32);
        bf.x = (lane < 16) ? t0 : p[sub][kc - 6];
        bf.y = (lane < 16) ? t1 : p[sub][kc - 5];
      }
      const int krow = kc4 * 4 + lhi * 2;
#pragma unroll
      for (int d = 0; d < 4; ++d) {
        v2f a;
        a.x = Vsb[swz(krow, d * 16 + l15)];
        a.y = Vsb[swz(krow + 1, d * 16 + l15)];
        o_acc[d] = WMMA_F32(a, bf, o_acc[d]);
      }
    }
  }

  // Normalize and store O[b, q, h, dh]
  const float inv_l = 1.0f / l_run;
#pragma unroll
  for (int d = 0; d < 4; ++d)
#pragma unroll
    for (int v = 0; v < 8; ++v) {
      int dh = d * 16 + v + lhi * 8;
      Ow[(((size_t)b * Lseq + qglob) * Hn + h) * 64 + dh] = o_acc[d][v] * inv_l;
    }
}

// ---------------------------------------------------------------------------
extern "C" void kernel_launch(void* const* d_in, const int* in_sizes, int n_in,
                              void* d_out, int out_size, void* d_ws,
                              size_t ws_size, hipStream_t stream) {
  (void)in_sizes; (void)n_in; (void)out_size; (void)ws_size;

  const int B = 2, L = 2048, D = 1024, H = 16;
  const int M = B * L;  // 4096

  const float* x  = (const float*)d_in[0];
  const float* Wq = (const float*)d_in[1];
  const float* bq = (const float*)d_in[2];
  const float* Wk = (const float*)d_in[3];
  const float* bk = (const float*)d_in[4];
  const float* Wv = (const float*)d_in[5];
  const float* bv = (const float*)d_in[6];
  const float* Wo = (const float*)d_in[7];
  const float* bo = (const float*)d_in[8];
  float* out = (float*)d_out;

  float* qws = (float*)d_ws;  // 4096*1024 floats each
  float* kws = qws + (size_t)M * D;
  float* vws = kws + (size_t)M * D;
  float* ows = vws + (size_t)M * D;

  dim3 gg(D / 64, M / 64);  // (16, 64)
  dim3 bb(128);

  gemm_bias_wmma_f32<<<gg, bb, 0, stream>>>(x, Wq, bq, qws, M, D, D, 1);
  gemm_bias_wmma_f32<<<gg, bb, 0, stream>>>(x, Wk, bk, kws, M, D, D, 0);
  gemm_bias_wmma_f32<<<gg, bb, 0, stream>>>(x, Wv, bv, vws, M, D, D, 0);

  flash_attn_wmma_f32<<<dim3(L / 64, H, B), bb, 0, stream>>>(qws, kws, vws,
                                                             ows, L, H);

  gemm_bias_wmma_f32<<<gg, bb, 0, stream>>>(ows, Wo, bo, out, M, D, D, 0);
}